// GATAutoEncoder_64544768524830
// MI455X (gfx1250) — compile-verified
//
#include <hip/hip_runtime.h>

#define NN 50000           // nodes (multiple of 16)
#define NE 800000          // raw edges
#define ET 850000          // edges incl. self loops
#define S  144             // padded feature stride (covers 129-wide layers)

typedef __attribute__((ext_vector_type(2))) float v2f;
typedef __attribute__((ext_vector_type(8))) float v8f;

__device__ __forceinline__ float lrelu(float v) { return v > 0.0f ? v : 0.2f * v; }

__device__ __forceinline__ void edge_sd(const int* __restrict__ ei, int e, int& s, int& d) {
  if (e < NE) { s = ei[e]; d = ei[NE + e]; }
  else        { s = e - NE; d = s; }      // appended self-loops
}

// float atomic max via signed/unsigned integer atomics (order-preserving encoding)
__device__ __forceinline__ void atomicMaxF(float* addr, float v) {
  if (v >= 0.0f) atomicMax((int*)addr, __float_as_int(v));
  else           atomicMin((unsigned int*)addr, __float_as_uint(v));
}

__global__ void pad_input_kernel(const float* __restrict__ x, float* __restrict__ A) {
  int tid = blockIdx.x * blockDim.x + threadIdx.x;
  if (tid >= NN * S) return;
  int col = tid % S, row = tid / S;
  A[tid] = (col < 6) ? x[row * 6 + col] : 0.0f;
}

__global__ void zero_kernel(float* __restrict__ p, int n) {
  int tid = blockIdx.x * blockDim.x + threadIdx.x;
  if (tid < n) p[tid] = 0.0f;
}

// Repack W[cin][HC] -> Wp[cin4/2][HCp][2] with zero padding so the GEMM B-fragment
// is one unconditional 8-byte load per lane (pair = (W[k][col], W[k+1][col])).
__global__ void pad_w_kernel(const float* __restrict__ W, float* __restrict__ Wp,
                             int cin, int cin4, int HC, int HCp) {
  int tid = blockIdx.x * blockDim.x + threadIdx.x;
  int total = (cin4 >> 1) * HCp;
  if (tid >= total) return;
  int col = tid % HCp, kp = tid / HCp;
  int k0 = kp * 2, k1 = k0 + 1;
  float v0 = (k0 < cin && col < HC) ? W[(size_t)k0 * HC + col] : 0.0f;
  float v1 = (k1 < cin && col < HC) ? W[(size_t)k1 * HC + col] : 0.0f;
  Wp[(size_t)tid * 2]     = v0;
  Wp[(size_t)tid * 2 + 1] = v1;
}

// XL[N, HCp] = A[N, cin(pad)] @ W; one wave = one 16x16 tile.
// A-fragment (32-bit A 16x4): lanes 0-15 hold K=k,k+1; lanes 16-31 hold K=k+2,k+3.
// B-fragment mirrors it via the pair-packed Wp layout. No guards in the K loop.
__global__ void gemm_wmma_kernel(const float* __restrict__ A, const float* __restrict__ Wp,
                                 float* __restrict__ XL, int cin4, int HCp) {
  const int lane    = threadIdx.x & 31;
  const int rowBase = blockIdx.x << 4;
  const int colBase = blockIdx.y << 4;
  const int m  = lane & 15;
  const int kb = (lane >> 4) << 1;     // 0 or 2
  const int col = colBase + m;
  const float* arow = A + (size_t)(rowBase + m) * S + kb;
  v8f c = {};
  for (int k = 0; k < cin4; k += 4) {
    v2f a = *(const v2f*)(arow + k);                                   // 8B load
    v2f b = *(const v2f*)(Wp + ((size_t)((k + kb) >> 1) * HCp + col) * 2); // 8B load
    c = __builtin_amdgcn_wmma_f32_16x16x4_f32(false, a, false, b, (short)0, c, false, false);
  }
  // C/D layout: VGPR r -> M = r (lanes 0-15) or 8+r (lanes 16-31), N = lane&15
  const int rOff = (lane >> 4) << 3;
  for (int r = 0; r < 8; ++r)
    XL[(size_t)(rowBase + rOff + r) * S + col] = c[r];   // zero-fills pad cols
}

// per (node, head): attention dots + init amax/denom
__global__ void att_init_kernel(const float* __restrict__ XL,
                                const float* __restrict__ att_src, const float* __restrict__ att_dst,
                                float* __restrict__ a_src, float* __restrict__ a_dst,
                                float* __restrict__ amax, float* __restrict__ denom,
                                int H, int C) {
  int tid = blockIdx.x * blockDim.x + threadIdx.x;
  if (tid >= NN * H) return;
  int n = tid / H, h = tid % H;
  const float* xr = XL + (size_t)n * S + h * C;
  const float* ws = att_src + h * C;
  const float* wd = att_dst + h * C;
  float ss = 0.0f, sd = 0.0f;
  for (int c = 0; c < C; ++c) { float v = xr[c]; ss += v * ws[c]; sd += v * wd[c]; }
  a_src[n * 4 + h] = ss;
  a_dst[n * 4 + h] = sd;
  amax [n * 4 + h] = -1e30f;
  denom[n * 4 + h] = 0.0f;
}

__global__ void edge_max_kernel(const int* __restrict__ ei,
                                const float* __restrict__ a_src, const float* __restrict__ a_dst,
                                float* __restrict__ amax, int H) {
  int e = blockIdx.x * blockDim.x + threadIdx.x;
  if (e >= ET) return;
  int s, d; edge_sd(ei, e, s, d);
  for (int h = 0; h < H; ++h) {
    float alpha = lrelu(a_src[s * 4 + h] + a_dst[d * 4 + h]);
    atomicMaxF(&amax[d * 4 + h], alpha);
  }
}

__global__ void edge_sum_kernel(const int* __restrict__ ei,
                                const float* __restrict__ a_src, const float* __restrict__ a_dst,
                                const float* __restrict__ amax, float* __restrict__ denom,
                                float* __restrict__ ae, int H) {
  int e = blockIdx.x * blockDim.x + threadIdx.x;
  if (e >= ET) return;
  int s, d; edge_sd(ei, e, s, d);
  for (int h = 0; h < H; ++h) {
    float alpha = lrelu(a_src[s * 4 + h] + a_dst[d * 4 + h]);
    float v = __expf(alpha - amax[d * 4 + h]);
    ae[(size_t)e * 4 + h] = v;
    atomicAdd(&denom[d * 4 + h], v);
  }
}

// 32 lanes per edge: coalesced gather of xl[src] + scatter-add into out[dst] (L2-resident)
__global__ void edge_aggr_kernel(const int* __restrict__ ei,
                                 const float* __restrict__ ae, const float* __restrict__ denom,
                                 const float* __restrict__ XL, float* __restrict__ OUT,
                                 int C, int HC) {
  int lane = threadIdx.x & 31;
  int e = (blockIdx.x * blockDim.x + threadIdx.x) >> 5;
  if (e >= ET) return;
  int s, d; edge_sd(ei, e, s, d);
  const float* xs = XL + (size_t)s * S;
  float*       od = OUT + (size_t)d * S;
  for (int col = lane; col < HC; col += 32) {
    int h = col / C;
    float w = ae[(size_t)e * 4 + h] / (denom[d * 4 + h] + 1e-16f);
    atomicAdd(&od[col], w * xs[col]);
  }
}

__global__ void bias_act_kernel(float* __restrict__ A, const float* __restrict__ bias,
                                int HC, int relu, float* __restrict__ cpy, int copyCols) {
  int tid = blockIdx.x * blockDim.x + threadIdx.x;
  if (tid >= NN * HC) return;
  int n = tid / HC, j = tid % HC;
  float v = A[(size_t)n * S + j] + bias[j];
  if (relu) v = lrelu(v);
  A[(size_t)n * S + j] = v;
  if (cpy && j < copyCols) cpy[(size_t)n * copyCols + j] = v;
}

extern "C" void kernel_launch(void* const* d_in, const int* in_sizes, int n_in,
                              void* d_out, int out_size, void* d_ws, size_t ws_size,
                              hipStream_t stream) {
  (void)in_sizes; (void)n_in; (void)out_size; (void)ws_size;
  const float* x  = (const float*)d_in[0];
  const int*   ei = (const int*)d_in[1];

  float* ws    = (float*)d_ws;
  float* A     = ws;                         // N*S  feature / accumulator buffer
  float* XL    = A + (size_t)NN * S;         // N*S  post-GEMM features
  float* a_src = XL + (size_t)NN * S;        // N*4
  float* a_dst = a_src + (size_t)NN * 4;     // N*4
  float* amax  = a_dst + (size_t)NN * 4;     // N*4
  float* denom = amax  + (size_t)NN * 4;     // N*4
  float* ae    = denom + (size_t)NN * 4;     // ET*4
  float* Wp    = ae + (size_t)ET * 4;        // <= 132/2 * 144 * 2 = 19008 floats
  float* out   = (float*)d_out;

  pad_input_kernel<<<(NN * S + 255) / 256, 256, 0, stream>>>(x, A);

  // layer configs: (cin, heads, C per head, H*C, relu-after)
  const int cins[6]  = {6, 129, 129, 128, 129, 129};
  const int Hs[6]    = {3, 3, 1, 3, 3, 1};
  const int Cs[6]    = {43, 43, 128, 43, 43, 6};
  const int HCs[6]   = {129, 129, 128, 129, 129, 6};
  const int relus[6] = {1, 1, 0, 1, 1, 0};

  for (int i = 0; i < 6; ++i) {
    const float* W  = (const float*)d_in[3 + 4 * i];
    const float* as = (const float*)d_in[4 + 4 * i];
    const float* ad = (const float*)d_in[5 + 4 * i];
    const float* bs = (const float*)d_in[6 + 4 * i];
    const int cin = cins[i], H = Hs[i], C = Cs[i], HC = HCs[i];
    const int cin4 = (cin + 3) & ~3;
    const int colTiles = (HC + 15) / 16;
    const int HCp = colTiles * 16;

    int wpn = (cin4 >> 1) * HCp;
    pad_w_kernel<<<(wpn + 255) / 256, 256, 0, stream>>>(W, Wp, cin, cin4, HC, HCp);

    dim3 gg(NN / 16, colTiles);
    gemm_wmma_kernel<<<gg, 32, 0, stream>>>(A, Wp, XL, cin4, HCp);

    att_init_kernel<<<(NN * H + 255) / 256, 256, 0, stream>>>(XL, as, ad, a_src, a_dst,
                                                              amax, denom, H, C);
    edge_max_kernel<<<(ET + 255) / 256, 256, 0, stream>>>(ei, a_src, a_dst, amax, H);
    edge_sum_kernel<<<(ET + 255) / 256, 256, 0, stream>>>(ei, a_src, a_dst, amax, denom, ae, H);

    zero_kernel<<<(NN * S + 255) / 256, 256, 0, stream>>>(A, NN * S);
    edge_aggr_kernel<<<((size_t)ET * 32 + 255) / 256, 256, 0, stream>>>(ei, ae, denom, XL, A, C, HC);

    float* cpy = nullptr; int cc = 0;
    if (i == 2) { cpy = out;                     cc = 128; }  // h
    if (i == 5) { cpy = out + (size_t)NN * 128;  cc = 6;   }  // h_recon
    bias_act_kernel<<<(NN * HC + 255) / 256, 256, 0, stream>>>(A, bs, HC, relus[i], cpy, cc);
  }
}